// GraphSAGE_79852031967993
// MI455X (gfx1250) — compile-verified
//
#include <hip/hip_runtime.h>

// GraphSAGE 2-layer forward for MI455X (gfx1250, wave32).
// Phase split: (a) edge scatter-mean via f32 global atomics (L2-resident agg),
// (b) fused [mean||feat] @ [W_l;W_r] GEMM using V_WMMA_F32_16X16X4_F32.
// Round 2: K-loop split at the concat boundary (mean-half -> W_l, feat-half ->
// W_r) so the inner loop is branch-free: ds_load_2addr + 2 global_load + wmma.

typedef __attribute__((ext_vector_type(2))) float v2f;
typedef __attribute__((ext_vector_type(8))) float v8f;

#define N_NODES 100000
#define N_EDGES 1600000
#define IN_DIM  128
#define HID_DIM 128
#define OUT_DIM 64

// ---------------------------------------------------------------- zero fill
__global__ void zero4_kernel(float4* __restrict__ p, int n4) {
    int i = blockIdx.x * blockDim.x + threadIdx.x;
    int stride = gridDim.x * blockDim.x;
    float4 z = make_float4(0.f, 0.f, 0.f, 0.f);
    for (; i < n4; i += stride) p[i] = z;
}

// ------------------------------------------------------------- edge scatter
// One wave32 per edge: 32 lanes x float4 = 128 features. Gather feat[src]
// (512 B coalesced), atomic-add into agg[dst] (51 MB -> lives in the 192 MB
// L2). Lane 0 counts in-degree (first pass only; deg == nullptr on layer 2).
__global__ __launch_bounds__(256)
void scatter_edges(const float* __restrict__ feat,
                   const int*   __restrict__ src,
                   const int*   __restrict__ dst,
                   float*       __restrict__ agg,
                   float*       __restrict__ deg,
                   int nEdges) {
    int wid  = (blockIdx.x * blockDim.x + threadIdx.x) >> 5;
    int lane = threadIdx.x & 31;
    if (wid >= nEdges) return;
    int s = src[wid];
    int d = dst[wid];
    float4 v = reinterpret_cast<const float4*>(feat + (size_t)s * IN_DIM)[lane];
    float* ap = agg + (size_t)d * IN_DIM + lane * 4;
    atomicAdd(ap + 0, v.x);
    atomicAdd(ap + 1, v.y);
    atomicAdd(ap + 2, v.z);
    atomicAdd(ap + 3, v.w);
    if (deg != nullptr && lane == 0) atomicAdd(deg + d, 1.0f);
}

// ------------------------------------------------------------- SAGE layer
// out[16 x NOUT] tile per block:  out = [mean || feat] @ [W_l ; W_r] + b.
// K = 256 total, consumed 4 at a time by v_wmma_f32_16x16x4_f32.
// A fragment (16x4 f32, ISA layout): lanes 0-15 -> M=lane,    K={k,k+1};
//                                    lanes 16-31 -> M=lane-16, K={k+2,k+3}.
// B fragment (4x16 f32): mirrored (lane half selects K pair, lane&15 = N).
// C/D: VGPR r -> M = r + (lane>=16 ? 8 : 0), N = lane&15.
template <int NOUT, bool RELU>
__global__ __launch_bounds__(32 * (NOUT / 16))
void sage_layer(const float* __restrict__ agg,
                const float* __restrict__ deg,
                const float* __restrict__ feat,
                const float* __restrict__ Wl,   // [128 x NOUT] row-major
                const float* __restrict__ Wr,   // [128 x NOUT] row-major
                const float* __restrict__ bias, // [NOUT]
                float*       __restrict__ out)  // [N x NOUT]
{
    constexpr int KTOT = 2 * IN_DIM; // 256
    __shared__ float As[KTOT][17];   // transposed [k][m]; stride 17 -> conflict-free

    const int tid     = threadIdx.x;
    const int rowBase = blockIdx.x * 16;

    // Fill LDS: rows 0..127 = mean = agg/deg, rows 128..255 = feat.
    for (int idx = tid; idx < 16 * 32; idx += blockDim.x) {
        int m  = idx >> 5;
        int d4 = idx & 31;
        int g  = rowBase + m;
        float4 xv = reinterpret_cast<const float4*>(feat + (size_t)g * IN_DIM)[d4];
        float4 av = reinterpret_cast<const float4*>(agg  + (size_t)g * IN_DIM)[d4];
        float dg  = deg[g];
        float inv = dg > 0.f ? 1.0f / dg : 0.f;
        int d = d4 * 4;
        As[d + 0][m] = av.x * inv;
        As[d + 1][m] = av.y * inv;
        As[d + 2][m] = av.z * inv;
        As[d + 3][m] = av.w * inv;
        As[IN_DIM + d + 0][m] = xv.x;
        As[IN_DIM + d + 1][m] = xv.y;
        As[IN_DIM + d + 2][m] = xv.z;
        As[IN_DIM + d + 3][m] = xv.w;
    }
    __syncthreads();

    const int wave  = tid >> 5;          // N-tile index
    const int lane  = tid & 31;
    const int m     = lane & 15;
    const int khalf = (lane >> 4) << 1;  // 0 or 2
    const int ncol  = wave * 16 + m;

    v8f acc = {};

    // ---- first half: mean @ W_l (K = 0..127), branch-free pointer walk ----
    {
        const float* wp = Wl + (size_t)khalf * NOUT + ncol;
#pragma unroll 4
        for (int k = 0; k < IN_DIM; k += 4) {
            v2f A;
            A.x = As[k + khalf + 0][m];
            A.y = As[k + khalf + 1][m];
            v2f B;
            B.x = wp[0];
            B.y = wp[NOUT];
            acc = __builtin_amdgcn_wmma_f32_16x16x4_f32(
                false, A, false, B, (short)0, acc, false, false);
            wp += 4 * NOUT;
        }
    }
    // ---- second half: feat @ W_r (K = 128..255) ----
    {
        const float* wp = Wr + (size_t)khalf * NOUT + ncol;
#pragma unroll 4
        for (int k = 0; k < IN_DIM; k += 4) {
            v2f A;
            A.x = As[IN_DIM + k + khalf + 0][m];
            A.y = As[IN_DIM + k + khalf + 1][m];
            v2f B;
            B.x = wp[0];
            B.y = wp[NOUT];
            acc = __builtin_amdgcn_wmma_f32_16x16x4_f32(
                false, A, false, B, (short)0, acc, false, false);
            wp += 4 * NOUT;
        }
    }

    const float bv = bias[ncol];
    const int mBase = rowBase + ((lane >> 4) << 3);
#pragma unroll
    for (int r = 0; r < 8; ++r) {
        float v = acc[r] + bv;
        if (RELU) v = v > 0.f ? v : 0.f;
        out[(size_t)(mBase + r) * NOUT + ncol] = v;
    }
}

// ---------------------------------------------------------------- launcher
extern "C" void kernel_launch(void* const* d_in, const int* in_sizes, int n_in,
                              void* d_out, int out_size, void* d_ws, size_t ws_size,
                              hipStream_t stream) {
    const float* x   = (const float*)d_in[0];
    const int*   ei  = (const int*)d_in[1];   // [2, E] int32
    const float* W1l = (const float*)d_in[2];
    const float* W1r = (const float*)d_in[3];
    const float* b1  = (const float*)d_in[4];
    const float* W2l = (const float*)d_in[5];
    const float* W2r = (const float*)d_in[6];
    const float* b2  = (const float*)d_in[7];
    float* out = (float*)d_out;

    const int E = in_sizes[1] / 2;
    const int* src = ei;
    const int* dst = ei + E;

    // Workspace layout: agg [N*128] | h [N*128] | deg [N]  (~103 MB)
    float* agg = (float*)d_ws;
    float* h   = agg + (size_t)N_NODES * HID_DIM;
    float* deg = h   + (size_t)N_NODES * HID_DIM;

    const int aggN4 = N_NODES * IN_DIM / 4;
    const int degN4 = N_NODES / 4;
    const int scatterBlocks = (E * 32 + 255) / 256; // 1 wave per edge, 8 waves/block
    const int rowBlocks = N_NODES / 16;             // 100000 / 16 = 6250 exact

    // ---- layer 1 ----
    zero4_kernel<<<2048, 256, 0, stream>>>((float4*)agg, aggN4);
    zero4_kernel<<<128, 256, 0, stream>>>((float4*)deg, degN4);
    scatter_edges<<<scatterBlocks, 256, 0, stream>>>(x, src, dst, agg, deg, E);
    sage_layer<HID_DIM, true><<<rowBlocks, 32 * (HID_DIM / 16), 0, stream>>>(
        agg, deg, x, W1l, W1r, b1, h);

    // ---- layer 2 ----
    zero4_kernel<<<2048, 256, 0, stream>>>((float4*)agg, aggN4);
    scatter_edges<<<scatterBlocks, 256, 0, stream>>>(h, src, dst, agg, nullptr, E);
    sage_layer<OUT_DIM, false><<<rowBlocks, 32 * (OUT_DIM / 16), 0, stream>>>(
        agg, deg, h, W2l, W2r, b2, out);
}